// GCN_88545045774432
// MI455X (gfx1250) — compile-verified
//
#include <hip/hip_runtime.h>

typedef __attribute__((ext_vector_type(2))) float v2f;
typedef __attribute__((ext_vector_type(8))) float v8f;

// ---------------------------------------------------------------------------
// GCN norm pipeline
// ---------------------------------------------------------------------------
__global__ void deg_init_k(float* __restrict__ deg, int N) {
  int t = blockIdx.x * blockDim.x + threadIdx.x;
  if (t < N) deg[t] = 1.0f;  // self-loop weight
}

__global__ void deg_scatter_k(const int* __restrict__ dst, const float* __restrict__ ew,
                              float* __restrict__ deg, int E) {
  int t = blockIdx.x * blockDim.x + threadIdx.x;
  if (t < E) unsafeAtomicAdd(&deg[dst[t]], ew[t]);
}

__global__ void dinv_k(const float* __restrict__ deg, float* __restrict__ dinv, int N) {
  int t = blockIdx.x * blockDim.x + threadIdx.x;
  if (t < N) {
    float d = deg[t];
    dinv[t] = (d > 0.0f) ? rsqrtf(d) : 0.0f;
  }
}

__global__ void norm_k(const int* __restrict__ src, const int* __restrict__ dst,
                       const float* __restrict__ ew, const float* __restrict__ dinv,
                       float* __restrict__ nrm, int E) {
  int t = blockIdx.x * blockDim.x + threadIdx.x;
  if (t < E) nrm[t] = dinv[src[t]] * ew[t] * dinv[dst[t]];
}

// ---------------------------------------------------------------------------
// Dense GEMM  H[N,F] = X[N,K] @ W[K,F]  via V_WMMA_F32_16X16X4_F32
// One wave computes one 16x16 output tile. EXEC stays all-ones per wave.
// A 16x4 layout: lane L holds row M=L%16, VGPRs = K pair selected by L/16.
// B 4x16 layout: VGPR r holds K=(r + 2*(L/16)), N = L%16.
// C/D 16x16  : VGPR r holds M=(r + 8*(L/16)), N = L%16.
// ---------------------------------------------------------------------------
template <int K, int F>
__global__ void gemm16_wmma(const float* __restrict__ X, const float* __restrict__ W,
                            float* __restrict__ H, int N) {
  const int lane = threadIdx.x & 31;
  const int wave = threadIdx.x >> 5;
  const int Ft = F >> 4;
  const int Mt = N >> 4;
  const int tile = blockIdx.x * (blockDim.x >> 5) + wave;
  if (tile >= Mt * Ft) return;          // wave-uniform exit
  const int mt = tile / Ft;
  const int ft = tile - mt * Ft;
  const int l16 = lane & 15;
  const int hi  = lane >> 4;            // 0 or 1

  const float* __restrict__ xrow = X + (size_t)(mt * 16 + l16) * K;
  const float* __restrict__ wcol = W + ft * 16 + l16;

  v8f acc = {};
#pragma unroll 4
  for (int k = 0; k < K; k += 4) {
    v2f a, b;
    a.x = xrow[k + 2 * hi];
    a.y = xrow[k + 2 * hi + 1];
    b.x = wcol[(size_t)(k + 2 * hi) * F];
    b.y = wcol[(size_t)(k + 2 * hi + 1) * F];
    acc = __builtin_amdgcn_wmma_f32_16x16x4_f32(false, a, false, b,
                                                (short)0, acc, false, false);
  }

  float* __restrict__ hcol = H + ft * 16 + l16;
#pragma unroll
  for (int r = 0; r < 8; ++r)
    hcol[(size_t)(mt * 16 + r + 8 * hi) * F] = acc[r];
}

// ---------------------------------------------------------------------------
// Aggregation: a[n,f] = bias[f] + dinv[n]^2 * h[n,f]   (self-loop, no atomics)
// then atomic scatter over edges: a[dst,f] += norm[e] * h[src,f]
// ---------------------------------------------------------------------------
template <int F>
__global__ void agg_init_k(const float* __restrict__ h, const float* __restrict__ dinv,
                           const float* __restrict__ bias, float* __restrict__ a, int N) {
  int t = blockIdx.x * blockDim.x + threadIdx.x;
  if (t >= N * F) return;
  constexpr int LOGF = (F == 64) ? 6 : 5;
  int n = t >> LOGF;
  int f = t & (F - 1);
  float di = dinv[n];
  a[t] = bias[f] + di * di * h[t];
}

template <int F>
__global__ void agg_edges_k(const float* __restrict__ h, const int* __restrict__ src,
                            const int* __restrict__ dst, const float* __restrict__ nrm,
                            float* __restrict__ a, int E) {
  unsigned t = blockIdx.x * blockDim.x + threadIdx.x;
  constexpr int LOGF = (F == 64) ? 6 : 5;
  unsigned e = t >> LOGF;
  if (e >= (unsigned)E) return;
  unsigned f = t & (F - 1);
  int s = src[e];                       // uniform across F consecutive lanes
  int d = dst[e];
  float w = nrm[e];
  unsafeAtomicAdd(&a[(size_t)d * F + f], w * h[(size_t)s * F + f]);
}

__global__ void relu_k(float* __restrict__ a, int n) {
  int t = blockIdx.x * blockDim.x + threadIdx.x;
  if (t < n) a[t] = fmaxf(a[t], 0.0f);
}

// ---------------------------------------------------------------------------
// Mean pool: one block per output column (F_out = 32)
// ---------------------------------------------------------------------------
__global__ void mean_pool_k(const float* __restrict__ emb, float* __restrict__ g, int N) {
  __shared__ float sm[256];
  const int c = blockIdx.x;             // 0..31
  float s = 0.0f;
  for (int r = threadIdx.x; r < N; r += blockDim.x) s += emb[(size_t)r * 32 + c];
  sm[threadIdx.x] = s;
  __syncthreads();
  for (int st = 128; st > 0; st >>= 1) {
    if (threadIdx.x < st) sm[threadIdx.x] += sm[threadIdx.x + st];
    __syncthreads();
  }
  if (threadIdx.x == 0) g[c] = sm[0] / (float)N;
}

// ---------------------------------------------------------------------------
// Launch
// ---------------------------------------------------------------------------
static inline unsigned cdiv(long a, long b) { return (unsigned)((a + b - 1) / b); }

extern "C" void kernel_launch(void* const* d_in, const int* in_sizes, int n_in,
                              void* d_out, int out_size, void* d_ws, size_t ws_size,
                              hipStream_t stream) {
  const float* x  = (const float*)d_in[0];
  const int*   ei = (const int*)d_in[1];
  const float* ew = (const float*)d_in[2];
  const float* W1 = (const float*)d_in[3];
  const float* b1 = (const float*)d_in[4];
  const float* W2 = (const float*)d_in[5];
  const float* b2 = (const float*)d_in[6];
  const float* W3 = (const float*)d_in[7];
  const float* b3 = (const float*)d_in[8];

  const int N = in_sizes[0] / 128;      // F_in = 128
  const int E = in_sizes[2];
  const int* src = ei;
  const int* dst = ei + E;

  float* ws   = (float*)d_ws;
  float* deg  = ws;  ws += N;
  float* dinv = ws;  ws += N;
  float* nrm  = ws;  ws += E;
  float* bufA = ws;  ws += (size_t)N * 64;
  float* bufB = ws;                     // N*64
  float* out  = (float*)d_out;          // [N*32 node emb | 32 graph emb]

  const dim3 blk(256);

  // GCN norm
  deg_init_k   <<<cdiv(N, 256), blk, 0, stream>>>(deg, N);
  deg_scatter_k<<<cdiv(E, 256), blk, 0, stream>>>(dst, ew, deg, E);
  dinv_k       <<<cdiv(N, 256), blk, 0, stream>>>(deg, dinv, N);
  norm_k       <<<cdiv(E, 256), blk, 0, stream>>>(src, dst, ew, dinv, nrm, E);

  // Layer 1: h1 = x @ W1 -> bufA ; a1 = agg(h1) -> bufB ; relu
  gemm16_wmma<128, 64><<<cdiv((long)(N / 16) * 4, 8), blk, 0, stream>>>(x, W1, bufA, N);
  agg_init_k<64> <<<cdiv((long)N * 64, 256), blk, 0, stream>>>(bufA, dinv, b1, bufB, N);
  agg_edges_k<64><<<cdiv((long)E * 64, 256), blk, 0, stream>>>(bufA, src, dst, nrm, bufB, E);
  relu_k         <<<cdiv((long)N * 64, 256), blk, 0, stream>>>(bufB, N * 64);

  // Layer 2: h2 = a1 @ W2 -> bufA ; a2 = agg(h2) -> bufB ; relu
  gemm16_wmma<64, 64><<<cdiv((long)(N / 16) * 4, 8), blk, 0, stream>>>(bufB, W2, bufA, N);
  agg_init_k<64> <<<cdiv((long)N * 64, 256), blk, 0, stream>>>(bufA, dinv, b2, bufB, N);
  agg_edges_k<64><<<cdiv((long)E * 64, 256), blk, 0, stream>>>(bufA, src, dst, nrm, bufB, E);
  relu_k         <<<cdiv((long)N * 64, 256), blk, 0, stream>>>(bufB, N * 64);

  // Layer 3: h3 = a2 @ W3 -> bufA (N x 32) ; agg -> d_out ; mean pool
  gemm16_wmma<64, 32><<<cdiv((long)(N / 16) * 2, 8), blk, 0, stream>>>(bufB, W3, bufA, N);
  agg_init_k<32> <<<cdiv((long)N * 32, 256), blk, 0, stream>>>(bufA, dinv, b3, out, N);
  agg_edges_k<32><<<cdiv((long)E * 32, 256), blk, 0, stream>>>(bufA, src, dst, nrm, out, E);
  mean_pool_k    <<<32, 256, 0, stream>>>(out, out + (size_t)N * 32, N);
}